// CartesianPerceiver_34952443855039
// MI455X (gfx1250) — compile-verified
//
#include <hip/hip_runtime.h>

// ---------------------------------------------------------------------------
// CartesianPerceiver for MI455X (gfx1250, wave32, WMMA).
// Dominant cost: streaming ctx_spa_embed (537MB) once -> HBM-bound (~23us).
// K/V proj, spa@k0 and attn@V mapped onto V_WMMA_F32_16X16X4_F32 tiles; small matmuls use VALU.
// ---------------------------------------------------------------------------

typedef float v2f __attribute__((ext_vector_type(2)));
typedef float v8f __attribute__((ext_vector_type(8)));

__device__ __forceinline__ v8f wmma_f32(v2f a, v2f b, v8f c) {
  // D = A(16x4) * B(4x16) + C(16x16), all f32
  return __builtin_amdgcn_wmma_f32_16x16x4_f32(false, a, false, b, (short)0, c,
                                               false, false);
}

// block-wide (128 threads) reductions via LDS
__device__ __forceinline__ float blk_sum(float v, float* red, int t) {
  red[t] = v; __syncthreads();
  for (int off = 64; off > 0; off >>= 1) {
    if (t < off) red[t] += red[t + off];
    __syncthreads();
  }
  float r = red[0]; __syncthreads();
  return r;
}
__device__ __forceinline__ float blk_max(float v, float* red, int t) {
  red[t] = v; __syncthreads();
  for (int off = 64; off > 0; off >>= 1) {
    if (t < off) red[t] = fmaxf(red[t], red[t + off]);
    __syncthreads();
  }
  float r = red[0]; __syncthreads();
  return r;
}
__device__ __forceinline__ float blk_min(float v, float* red, int t) {
  red[t] = v; __syncthreads();
  for (int off = 64; off > 0; off >>= 1) {
    if (t < off) red[t] = fminf(red[t], red[t + off]);
    __syncthreads();
  }
  float r = red[0]; __syncthreads();
  return r;
}

// ---------------------------------------------------------------------------
// Kernel 1: per-batch gbias + mask -> additive score offset
// offs[b][s] = -1e9*mask - (d2-min)/(max-min+1e-6)*bias_factor
// ---------------------------------------------------------------------------
__global__ void precompute_kernel(const int* __restrict__ mask,
                                  const float* __restrict__ dists,
                                  const float* __restrict__ bias_factor,
                                  float* __restrict__ offs) {
  __shared__ float red[128];
  int b = blockIdx.x, t = threadIdx.x;
  float mn = 3.4e38f, mx = -3.4e38f;
  for (int s = t; s < 1024; s += 128) {
    float d = dists[b * 1024 + s];
    float d2 = d * d;
    mn = fminf(mn, d2);
    mx = fmaxf(mx, d2);
  }
  float gmn = blk_min(mn, red, t);
  float gmx = blk_max(mx, red, t);
  float inv = bias_factor[0] / (gmx - gmn + 1e-6f);
  for (int s = t; s < 1024; s += 128) {
    float d = dists[b * 1024 + s];
    float d2 = d * d;
    float g = (d2 - gmn) * inv;
    float mk = mask[b * 1024 + s] ? -1e9f : 0.f;
    offs[b * 1024 + s] = mk - g;
  }
}

// ---------------------------------------------------------------------------
// Kernel 2: layer-0 latent-query projection + q_spa transform (+1/8 scale).
// qprime[b][h][qi][i] = 0.125 * sum_j qs[b][i][j]*qhead[h][qi][j]
// ---------------------------------------------------------------------------
__global__ void qproj_kernel(const float* __restrict__ latent_q,
                             const float* __restrict__ proj_w,
                             const float* __restrict__ q_spa,
                             float* __restrict__ qprime) {
  __shared__ float xab[2][4][64];  // [pair][qi][64]
  __shared__ float qh0[4][4][64];  // [h][qi][64]
  int b = blockIdx.x, t = threadIdx.x;
  for (int i = t; i < 512; i += 128) {
    int pair = i >> 8, rem = i & 255, qi = rem >> 6, o = rem & 63;
    const float* W = proj_w + pair * 128 * 64;               // pw[0][0],[0][1]
    const float* L = latent_q + (pair * 4 + qi) * 128;
    float acc = 0.f;
    for (int in = 0; in < 128; ++in) acc += L[in] * W[in * 64 + o];
    xab[pair][qi][o] = acc;
  }
  __syncthreads();
  for (int i = t; i < 1024; i += 128) {
    int h = i >> 8, qi = (i >> 6) & 3, j = i & 63;
    qh0[h][qi][j] = (j < 32) ? xab[0][qi][(h & 1) * 32 + j]
                             : xab[1][qi][(h >> 1) * 32 + j - 32];
  }
  __syncthreads();
  for (int i = t; i < 1024; i += 128) {
    int h = i >> 8, qi = (i >> 6) & 3, ii = i & 63;
    const float* qsrow = q_spa + b * 4096 + ii * 64;
    float acc = 0.f;
    for (int j = 0; j < 64; ++j) acc += qsrow[j] * qh0[h][qi][j];
    qprime[((b * 4 + h) * 4 + qi) * 64 + ii] = acc * 0.125f;
  }
}

// ---------------------------------------------------------------------------
// Kernel 3: K/V projections via WMMA GEMM, scatter into head layout.
// Y(32768x128) = X(32768x128) @ [Wk|Wv](128x128) per embed; each output
// element feeds two heads.  grid = 2 embeds * 512 Mblocks, block = 4 waves.
// ---------------------------------------------------------------------------
__global__ void kvproj_kernel(const float* __restrict__ a_embed,
                              const float* __restrict__ b_embed,
                              const float* __restrict__ proj_w,
                              float* __restrict__ k0, float* __restrict__ v0) {
  int blk = blockIdx.x;
  int e = blk >> 9;
  int mblk = blk & 511;
  int lane = threadIdx.x & 31;
  int wave = threadIdx.x >> 5;
  int m0 = mblk * 64 + wave * 16;
  const float* X = (e == 0) ? a_embed : b_embed;
  const float* Wk = proj_w + (2 + e) * 128 * 64;  // pw[0][2+e]
  const float* Wv = proj_w + (4 + e) * 128 * 64;  // pw[0][4+e]
  int mrow = lane & 15;
  int khalf = (lane >> 4) * 2;

  const float* Wcol[8];
  for (int nt = 0; nt < 8; ++nt) {
    int c = nt * 16 + mrow;
    Wcol[nt] = (c < 64) ? (Wk + c) : (Wv + (c - 64));
  }
  v8f zero = {};
  v8f acc[8];
  for (int nt = 0; nt < 8; ++nt) acc[nt] = zero;

  const float* xrow = X + (m0 + mrow) * 128;
  for (int k = 0; k < 128; k += 4) {
    v2f a;
    a.x = xrow[k + khalf + 0];
    a.y = xrow[k + khalf + 1];
    for (int nt = 0; nt < 8; ++nt) {
      v2f bb;
      bb.x = Wcol[nt][(k + khalf + 0) * 64];
      bb.y = Wcol[nt][(k + khalf + 1) * 64];
      acc[nt] = wmma_f32(a, bb, acc[nt]);
    }
  }
  int rbase = (lane >> 4) * 8;
  for (int nt = 0; nt < 8; ++nt) {
    int c = nt * 16 + mrow;
    int part = c >> 6;  // 0=K, 1=V
    int cc = c & 63;
    int half = cc >> 5;
    int d = cc & 31;
    float* dst = (part == 0) ? k0 : v0;
    int h1, h2, doff;
    if (e == 0) { h1 = half; h2 = half + 2; doff = d; }
    else        { h1 = half * 2; h2 = half * 2 + 1; doff = 32 + d; }
    for (int r = 0; r < 8; ++r) {
      int row = m0 + rbase + r;
      int b_ = row >> 10, s_ = row & 1023;
      float val = acc[nt][r];
      dst[((b_ * 4 + h1) * 1024 + s_) * 64 + doff] = val;
      dst[((b_ * 4 + h2) * 1024 + s_) * 64 + doff] = val;
    }
  }
}

// ---------------------------------------------------------------------------
// Kernel 4: THE streaming kernel.  Per (b,s): k'[i][h] = spa[b,s] @ k0 via
// WMMA (M=i 4 tiles, K=j 16 steps, N=heads 4-of-16), then fused score dot
// with qprime.  537MB of spa streamed exactly once; prefetch next tile.
// grid = 32 b * 32 s-chunks, block = 128 (4 waves, 1 s per wave per iter).
// ---------------------------------------------------------------------------
__global__ void spa_scores_kernel(const float* __restrict__ ctx_spa,
                                  const float* __restrict__ k0,
                                  const float* __restrict__ qprime,
                                  const float* __restrict__ offs,
                                  float* __restrict__ scores) {
  __shared__ float qp[4 * 4 * 64];   // [h][qi][i]
  __shared__ float kp[4][64][4];     // [wave][i][h]
  int b = blockIdx.x >> 5;
  int sc = blockIdx.x & 31;
  int t = threadIdx.x;
  int lane = t & 31, wave = t >> 5;
  int mrow = lane & 15, khalf = (lane >> 4) * 2;
  for (int i = t; i < 1024; i += 128) qp[i] = qprime[b * 1024 + i];
  __syncthreads();

  for (int iter = 0; iter < 8; ++iter) {
    int s = sc * 32 + iter * 4 + wave;
    const float* spa = ctx_spa + (size_t)(b * 1024 + s) * 4096;
    if (iter + 1 < 8) {
      const float* nspa = ctx_spa + (size_t)(b * 1024 + s + 4) * 4096;
      for (int pf = 0; pf < 4; ++pf)
        __builtin_prefetch(nspa + lane * 32 + pf * 1024, 0, 1);
    }
    v8f zero = {};
    v8f acc[4];
    for (int mt = 0; mt < 4; ++mt) acc[mt] = zero;
    bool bvalid = (mrow < 4);
    const float* krow = k0 + ((b * 4 + (mrow & 3)) * 1024 + s) * 64;
    for (int k = 0; k < 64; k += 4) {
      v2f bb;
      if (bvalid) { bb.x = krow[k + khalf + 0]; bb.y = krow[k + khalf + 1]; }
      else        { bb.x = 0.f; bb.y = 0.f; }
      for (int mt = 0; mt < 4; ++mt) {
        const float* arow = spa + (mt * 16 + mrow) * 64 + k + khalf;
        v2f aa; aa.x = arow[0]; aa.y = arow[1];
        acc[mt] = wmma_f32(aa, bb, acc[mt]);
      }
    }
    if (mrow < 4) {
      int rb = (lane >> 4) * 8;
      for (int mt = 0; mt < 4; ++mt)
        for (int r = 0; r < 8; ++r)
          kp[wave][mt * 16 + rb + r][mrow] = acc[mt][r];
    }
    __syncthreads();
    if (t < 64) {
      int sl = t >> 4, rem = t & 15, h = rem >> 2, qi = rem & 3;
      const float* q = qp + (h * 4 + qi) * 64;
      float sum = 0.f;
      for (int i = 0; i < 64; ++i) sum += q[i] * kp[sl][i][h];
      int ss = sc * 32 + iter * 4 + sl;
      scores[((b * 4 + h) * 4 + qi) * 1024 + ss] = sum + offs[b * 1024 + ss];
    }
    __syncthreads();
  }
}

// ---------------------------------------------------------------------------
// Kernel 5: softmax over 1024 keys (4 rows) + attn@V via WMMA.
// grid = 128 (b,h) blocks, block = 128 (4 waves, one 16-col N-tile each).
// ---------------------------------------------------------------------------
__global__ void softmax_av_kernel(const float* __restrict__ scores,
                                  const float* __restrict__ v0,
                                  float* __restrict__ out0) {
  __shared__ float attn[4 * 1024];
  __shared__ float red[128];
  int bh = blockIdx.x;
  int t = threadIdx.x;
  const float* srow = scores + bh * 4096;
  for (int qi = 0; qi < 4; ++qi) {
    float m = -3.4e38f;
    for (int s = t; s < 1024; s += 128) m = fmaxf(m, srow[qi * 1024 + s]);
    float mqi = blk_max(m, red, t);
    float lsum = 0.f;
    for (int s = t; s < 1024; s += 128) {
      float e = __expf(srow[qi * 1024 + s] - mqi);
      attn[qi * 1024 + s] = e;
      lsum += e;
    }
    float inv = 1.f / blk_sum(lsum, red, t);
    for (int s = t; s < 1024; s += 128) attn[qi * 1024 + s] *= inv;
  }
  __syncthreads();

  int lane = t & 31, wave = t >> 5;
  int mrow = lane & 15, khalf = (lane >> 4) * 2;
  int n0 = wave * 16;
  const float* vbase = v0 + (size_t)bh * 1024 * 64;
  v8f acc = {};
  for (int k = 0; k < 1024; k += 4) {
    v2f aa;
    if (mrow < 4) {
      aa.x = attn[mrow * 1024 + k + khalf + 0];
      aa.y = attn[mrow * 1024 + k + khalf + 1];
    } else { aa.x = 0.f; aa.y = 0.f; }
    v2f bb;
    bb.x = vbase[(k + khalf + 0) * 64 + n0 + mrow];
    bb.y = vbase[(k + khalf + 1) * 64 + n0 + mrow];
    acc = wmma_f32(aa, bb, acc);
  }
  if (lane < 16) {
    for (int r = 0; r < 4; ++r)
      out0[(bh * 4 + r) * 64 + n0 + lane] = acc[r];
  }
}

// ---------------------------------------------------------------------------
// Kernel 6: layer-0 epilogue: wo projection, residual+LN, FFD, residual+LN.
// grid = 32 batches, block = 128 (one thread per output column).
// ---------------------------------------------------------------------------
__global__ void post0_kernel(const float* __restrict__ out0,
                             const float* __restrict__ wo,
                             const float* __restrict__ latent_q,
                             const float* __restrict__ ffd_w,
                             const float* __restrict__ ffd_b,
                             const float* __restrict__ ln_scale,
                             const float* __restrict__ ln_bias,
                             float* __restrict__ emb1) {
  __shared__ float vin[4][128];
  __shared__ float xln[4][128];
  __shared__ float red[128];
  int b = blockIdx.x, t = threadIdx.x;
  for (int p = 0; p < 2; ++p) {
    for (int j = 0; j < 4; ++j) {
      int idx = t + 128 * j;
      int qi = idx >> 7, m = idx & 127, h = m >> 5, c = m & 31;
      vin[qi][m] = out0[((b * 4 + h) * 4 + qi) * 64 + p * 32 + c];
    }
    __syncthreads();
    const float* W = wo + p * 128 * 128;      // wo[0][p]
    for (int qi = 0; qi < 4; ++qi) {
      float ao = 0.f;
      for (int m = 0; m < 128; ++m) ao += vin[qi][m] * W[m * 128 + t];
      float x = latent_q[(p * 4 + qi) * 128 + t] + ao;
      float mean = blk_sum(x, red, t) * (1.f / 128.f);
      float d = x - mean;
      float var = blk_sum(d * d, red, t) * (1.f / 128.f);
      xln[qi][t] = d * rsqrtf(var + 1e-6f) * ln_scale[p * 128 + t] +
                   ln_bias[p * 128 + t];
      __syncthreads();
    }
    const float* Wf = ffd_w + p * 128 * 128;  // ffd_w[0][p]
    for (int qi = 0; qi < 4; ++qi) {
      float acc = 0.f;
      for (int m = 0; m < 128; ++m) acc += xln[qi][m] * Wf[m * 128 + t];
      float y = xln[qi][t] + acc + ffd_b[p * 128 + t];
      float mean = blk_sum(y, red, t) * (1.f / 128.f);
      float d = y - mean;
      float var = blk_sum(d * d, red, t) * (1.f / 128.f);
      emb1[((p * 32 + b) * 4 + qi) * 128 + t] =
          d * rsqrtf(var + 1e-6f) * ln_scale[(2 + p) * 128 + t] +
          ln_bias[(2 + p) * 128 + t];
      __syncthreads();
    }
    __syncthreads();
  }
}

// ---------------------------------------------------------------------------
// Kernel 7: final layer (ql=1, sl=4, no mask/bias, keys transformed by q_spa).
// Tiny -> one block per batch, pure VALU.
// ---------------------------------------------------------------------------
__global__ void final_kernel(const float* __restrict__ q_a,
                             const float* __restrict__ q_b,
                             const float* __restrict__ q_spa,
                             const float* __restrict__ emb1,
                             const float* __restrict__ proj_w,
                             const float* __restrict__ wo,
                             const float* __restrict__ ffd_w,
                             const float* __restrict__ ffd_b,
                             const float* __restrict__ ln_scale,
                             const float* __restrict__ ln_bias,
                             float* __restrict__ d_out) {
  __shared__ float qs[4096];
  __shared__ float e_lds[2][4][128];
  __shared__ float qv[2][128];
  __shared__ float xq[128];
  __shared__ float qh[4][64];
  __shared__ float qp2[4][64];
  __shared__ float xk[2][4][64];
  __shared__ float xv[2][4][64];
  __shared__ float kh[4][4][64];
  __shared__ float vh[4][4][64];
  __shared__ float kp2[4][4][64];
  __shared__ float sc2[4][4];
  __shared__ float at2[4][4];
  __shared__ float outh[4][64];
  __shared__ float vin2[2][128];
  __shared__ float xln2[128];
  __shared__ float red[128];
  int b = blockIdx.x, t = threadIdx.x;

  for (int i = t; i < 4096; i += 128) qs[i] = q_spa[b * 4096 + i];
  for (int i = t; i < 1024; i += 128) {
    int p = i >> 9, r = i & 511;
    e_lds[p][r >> 7][r & 127] = emb1[(p * 32 + b) * 512 + r];
  }
  qv[0][t] = q_a[b * 128 + t];
  qv[1][t] = q_b[b * 128 + t];
  __syncthreads();

  {  // q projection (pw[1][0], pw[1][1])
    int pair = t >> 6, o = t & 63;
    const float* W = proj_w + (6 + pair) * 128 * 64;
    float s = 0.f;
    for (int in = 0; in < 128; ++in) s += qv[pair][in] * W[in * 64 + o];
    xq[t] = s;
  }
  __syncthreads();
  for (int i = t; i < 256; i += 128) {
    int h = i >> 6, d = i & 63;
    qh[h][d] = (d < 32) ? xq[(h & 1) * 32 + d]
                        : xq[64 + (h >> 1) * 32 + (d - 32)];
  }
  __syncthreads();
  for (int i = t; i < 256; i += 128) {
    int h = i >> 6, ii = i & 63;
    float s = 0.f;
    for (int j = 0; j < 64; ++j) s += qs[ii * 64 + j] * qh[h][j];
    qp2[h][ii] = s * 0.125f;
  }
  // k/v projections (pw[1][2..5])
  for (int i = t; i < 1024; i += 128) {
    int which = i >> 8;  // 0,1 -> K(a,b); 2,3 -> V(a,b)
    int rem = i & 255, s_ = rem >> 6, o = rem & 63;
    int pair = which & 1;
    int j = (which < 2) ? (2 + pair) : (4 + pair);
    const float* W = proj_w + (6 + j) * 128 * 64;
    float acc = 0.f;
    for (int in = 0; in < 128; ++in) acc += e_lds[pair][s_][in] * W[in * 64 + o];
    if (which < 2) xk[pair][s_][o] = acc; else xv[pair][s_][o] = acc;
  }
  __syncthreads();
  for (int i = t; i < 2048; i += 128) {
    int kv = i >> 10, rem = i & 1023;
    int h = rem >> 8, s_ = (rem >> 6) & 3, d = rem & 63;
    float val;
    if (d < 32) val = (kv ? xv : xk)[0][s_][(h & 1) * 32 + d];
    else        val = (kv ? xv : xk)[1][s_][(h >> 1) * 32 + (d - 32)];
    if (kv) vh[h][s_][d] = val; else kh[h][s_][d] = val;
  }
  __syncthreads();
  for (int i = t; i < 1024; i += 128) {
    int h = i >> 8, s_ = (i >> 6) & 3, ii = i & 63;
    float acc = 0.f;
    for (int j = 0; j < 64; ++j) acc += qs[ii * 64 + j] * kh[h][s_][j];
    kp2[h][s_][ii] = acc;
  }
  __syncthreads();
  if (t < 16) {
    int h = t >> 2, s_ = t & 3;
    float acc = 0.f;
    for (int ii = 0; ii < 64; ++ii) acc += qp2[h][ii] * kp2[h][s_][ii];
    sc2[h][s_] = acc;
  }
  __syncthreads();
  if (t < 4) {
    int h = t;
    float m = fmaxf(fmaxf(sc2[h][0], sc2[h][1]), fmaxf(sc2[h][2], sc2[h][3]));
    float e0 = __expf(sc2[h][0] - m), e1 = __expf(sc2[h][1] - m);
    float e2 = __expf(sc2[h][2] - m), e3 = __expf(sc2[h][3] - m);
    float inv = 1.f / (e0 + e1 + e2 + e3);
    at2[h][0] = e0 * inv; at2[h][1] = e1 * inv;
    at2[h][2] = e2 * inv; at2[h][3] = e3 * inv;
  }
  __syncthreads();
  for (int i = t; i < 256; i += 128) {
    int h = i >> 6, d = i & 63;
    float acc = 0.f;
    for (int s_ = 0; s_ < 4; ++s_) acc += at2[h][s_] * vh[h][s_][d];
    outh[h][d] = acc;
  }
  __syncthreads();
  for (int i = t; i < 256; i += 128) {
    int p = i >> 7, m = i & 127, h = m >> 5, c = m & 31;
    vin2[p][m] = outh[h][p * 32 + c];
  }
  __syncthreads();
  for (int p = 0; p < 2; ++p) {
    const float* W = wo + (2 + p) * 128 * 128;   // wo[1][p]
    float ao = 0.f;
    for (int m = 0; m < 128; ++m) ao += vin2[p][m] * W[m * 128 + t];
    float x = qv[p][t] + ao;
    float mean = blk_sum(x, red, t) * (1.f / 128.f);
    float d = x - mean;
    float var = blk_sum(d * d, red, t) * (1.f / 128.f);
    float xn = d * rsqrtf(var + 1e-6f) * ln_scale[(4 + p) * 128 + t] +
               ln_bias[(4 + p) * 128 + t];
    xln2[t] = xn;
    __syncthreads();
    const float* Wf = ffd_w + (2 + p) * 128 * 128;  // ffd_w[1][p]
    float fa = 0.f;
    for (int m = 0; m < 128; ++m) fa += xln2[m] * Wf[m * 128 + t];
    float y = xn + fa + ffd_b[(2 + p) * 128 + t];
    mean = blk_sum(y, red, t) * (1.f / 128.f);
    d = y - mean;
    var = blk_sum(d * d, red, t) * (1.f / 128.f);
    d_out[(p * 32 + b) * 128 + t] =
        d * rsqrtf(var + 1e-6f) * ln_scale[(6 + p) * 128 + t] +
        ln_bias[(6 + p) * 128 + t];
    __syncthreads();
  }
}

// ---------------------------------------------------------------------------
extern "C" void kernel_launch(void* const* d_in, const int* in_sizes, int n_in,
                              void* d_out, int out_size, void* d_ws,
                              size_t ws_size, hipStream_t stream) {
  const float* a_embed     = (const float*)d_in[0];
  const float* b_embed     = (const float*)d_in[1];
  const float* q_a_embed   = (const float*)d_in[2];
  const float* q_b_embed   = (const float*)d_in[3];
  const float* ctx_spa     = (const float*)d_in[4];
  const float* q_spa       = (const float*)d_in[5];
  const int*   mask        = (const int*)d_in[6];
  const float* dists       = (const float*)d_in[7];
  const float* proj_w      = (const float*)d_in[8];
  const float* wo          = (const float*)d_in[9];
  const float* ffd_w       = (const float*)d_in[10];
  const float* ffd_b       = (const float*)d_in[11];
  const float* ln_scale    = (const float*)d_in[12];
  const float* ln_bias     = (const float*)d_in[13];
  const float* latent_q    = (const float*)d_in[14];
  const float* bias_factor = (const float*)d_in[15];
  float* out = (float*)d_out;

  float* ws      = (float*)d_ws;
  float* offs    = ws;                    // 32*1024
  float* qprime  = offs + 32768;          // 32*4*4*64
  float* k0      = qprime + 32768;        // 32*4*1024*64
  float* v0      = k0 + 8388608;          // 32*4*1024*64
  float* scores  = v0 + 8388608;          // 32*4*4*1024
  float* out0    = scores + 524288;       // 32*4*4*64
  float* emb1    = out0 + 32768;          // 2*32*4*128
  (void)in_sizes; (void)n_in; (void)out_size; (void)ws_size;

  precompute_kernel<<<32, 128, 0, stream>>>(mask, dists, bias_factor, offs);
  qproj_kernel<<<32, 128, 0, stream>>>(latent_q, proj_w, q_spa, qprime);
  kvproj_kernel<<<1024, 128, 0, stream>>>(a_embed, b_embed, proj_w, k0, v0);
  spa_scores_kernel<<<1024, 128, 0, stream>>>(ctx_spa, k0, qprime, offs, scores);
  softmax_av_kernel<<<128, 128, 0, stream>>>(scores, v0, out0);
  post0_kernel<<<32, 128, 0, stream>>>(out0, wo, latent_q, ffd_w, ffd_b,
                                       ln_scale, ln_bias, emb1);
  final_kernel<<<32, 128, 0, stream>>>(q_a_embed, q_b_embed, q_spa, emb1,
                                       proj_w, wo, ffd_w, ffd_b, ln_scale,
                                       ln_bias, out);
}